// GraphConvLayer_7086696039036
// MI455X (gfx1250) — compile-verified
//
#include <hip/hip_runtime.h>
#include <hip/hip_bf16.h>

typedef __attribute__((ext_vector_type(2))) float v2f;
typedef __attribute__((ext_vector_type(8))) float v8f;

#define IN_F 64
#define OUT_F 64
#define NSUP 3
#define LDS_STRIDE 68   // 68 % 64 == 4 -> conflict-free b64 reads (see analysis)

// ---------------------------------------------------------------------------
// Kernel 1: Y[k] = X @ W_k  via V_WMMA_F32_16X16X4_F32.
// Block = 256 threads (8 waves). Each block stages W_k^T in LDS once; each
// wave computes a full 16x64 row-block of Y[k] with 4 accumulators, so the
// A-operand (X) is loaded exactly once. 64 WMMA per wave.
// grid.y = support k, grid.x covers mTiles/8 blocks.
// ---------------------------------------------------------------------------
__global__ void xw_wmma_kernel(const float* __restrict__ X,
                               const float* __restrict__ W,
                               float* __restrict__ Y,
                               int nNodes, int mTiles) {
    __shared__ float WT[OUT_F * LDS_STRIDE];   // W_k transposed: WT[c][r]

    const int t    = (int)threadIdx.x;
    const int lane = t & 31;
    const int wave = t >> 5;
    const int k    = (int)blockIdx.y;

    // --- cooperative transpose fill: WT[c*68 + r] = W_k[r][c] (16KB) ---
    const float* __restrict__ Wk = W + (size_t)k * IN_F * OUT_F;
    {
        const int r  = t >> 2;             // 0..63
        const int cb = (t & 3) * 16;       // 0,16,32,48
#pragma unroll
        for (int j = 0; j < 16; ++j) {
            WT[(cb + j) * LDS_STRIDE + r] = Wk[(size_t)r * OUT_F + cb + j];
        }
    }
    __syncthreads();

    const int mTile = (int)blockIdx.x * 8 + wave;
    if (mTile >= mTiles) return;           // wave-uniform, after the barrier

    const int m0    = mTile * 16;
    const int mLane = lane & 15;
    const int kOff  = (lane >> 4) * 2;     // A/B lane mapping: K split 0/2
    const float* __restrict__ aRow = X + (size_t)(m0 + mLane) * IN_F + kOff;

    v8f acc0 = {}, acc1 = {}, acc2 = {}, acc3 = {};
#pragma unroll
    for (int kk = 0; kk < IN_F; kk += 4) {
        v2f a  = *(const v2f*)(aRow + kk);                                // b64
        v2f b0 = *(const v2f*)(&WT[( 0 + mLane) * LDS_STRIDE + kk + kOff]);
        v2f b1 = *(const v2f*)(&WT[(16 + mLane) * LDS_STRIDE + kk + kOff]);
        v2f b2 = *(const v2f*)(&WT[(32 + mLane) * LDS_STRIDE + kk + kOff]);
        v2f b3 = *(const v2f*)(&WT[(48 + mLane) * LDS_STRIDE + kk + kOff]);
        acc0 = __builtin_amdgcn_wmma_f32_16x16x4_f32(false, a, false, b0,
                                                     (short)0, acc0, false, false);
        acc1 = __builtin_amdgcn_wmma_f32_16x16x4_f32(false, a, false, b1,
                                                     (short)0, acc1, false, false);
        acc2 = __builtin_amdgcn_wmma_f32_16x16x4_f32(false, a, false, b2,
                                                     (short)0, acc2, false, false);
        acc3 = __builtin_amdgcn_wmma_f32_16x16x4_f32(false, a, false, b3,
                                                     (short)0, acc3, false, false);
    }

    // C/D layout: VGPR v -> row m0+v (lanes 0-15) / m0+8+v (lanes 16-31)
    float* __restrict__ Yk = Y + (size_t)k * nNodes * OUT_F;
    const int rBase = m0 + ((lane >> 4) * 8);
#pragma unroll
    for (int v = 0; v < 8; ++v) {
        float* __restrict__ orow = Yk + (size_t)(rBase + v) * OUT_F;
        orow[ 0 + mLane] = acc0[v];
        orow[16 + mLane] = acc1[v];
        orow[32 + mLane] = acc2[v];
        orow[48 + mLane] = acc3[v];
    }
}

// ---------------------------------------------------------------------------
// Kernel 2: out[i][f] = bias[f]   (also un-poisons d_out)
// ---------------------------------------------------------------------------
__global__ void init_bias_kernel(float* __restrict__ out,
                                 const float* __restrict__ bias, int n) {
    int i = (int)(blockIdx.x * blockDim.x + threadIdx.x);
    if (i < n) out[i] = bias[i & (OUT_F - 1)];
}

// ---------------------------------------------------------------------------
// Kernel 3: out[row] += val * Y[k][col] for every edge. One wave per edge;
// each lane handles 2 floats (b64 gather + 2 global_atomic_add_f32). Y
// (76.8MB) and out (25.6MB) are L2-resident (192MB L2), so the random
// gather/atomics are served by L2, not HBM.
// ---------------------------------------------------------------------------
__global__ void spmm_scatter_kernel(const int* __restrict__ rows,
                                    const int* __restrict__ cols,
                                    const float* __restrict__ vals,
                                    const float* __restrict__ Y,
                                    float* __restrict__ out,
                                    int E, int nNodes) {
    const long long tid = (long long)blockIdx.x * blockDim.x + threadIdx.x;
    const long long e   = tid >> 5;
    const int lane      = (int)(threadIdx.x & 31);
    if (e >= E) return;

    const int k = (int)blockIdx.y;
    const size_t eIdx = (size_t)k * E + (size_t)e;

    const int   r = rows[eIdx];
    const int   c = cols[eIdx];
    const float v = vals[eIdx];

    const float* __restrict__ y = Y + ((size_t)k * nNodes + (size_t)c) * OUT_F
                                    + lane * 2;
    v2f d = *(const v2f*)y;

    float* __restrict__ o = out + (size_t)r * OUT_F + lane * 2;
    atomicAdd(o,     v * d.x);
    atomicAdd(o + 1, v * d.y);
}

extern "C" void kernel_launch(void* const* d_in, const int* in_sizes, int n_in,
                              void* d_out, int out_size, void* d_ws, size_t ws_size,
                              hipStream_t stream) {
    const float* X    = (const float*)d_in[0];   // [N, 64]
    const int*   rows = (const int*)  d_in[1];   // [3, E]
    const int*   cols = (const int*)  d_in[2];   // [3, E]
    const float* vals = (const float*)d_in[3];   // [3, E]
    const float* W    = (const float*)d_in[4];   // [192, 64]
    const float* bias = (const float*)d_in[5];   // [64]

    const int nNodes = in_sizes[0] / IN_F;
    const int E      = in_sizes[1] / NSUP;

    float* Y   = (float*)d_ws;                   // [3, N, 64] = 76.8 MB
    float* out = (float*)d_out;                  // [N, 64]

    // 1) Dense GEMMs via WMMA: Y[k] = X @ W_k  (out = b + sum_k S_k (X W_k))
    const int mTiles = (nNodes + 15) / 16;       // 6250
    dim3 gGrid((mTiles + 7) / 8, NSUP);          // 8 m-tiles (waves) per block
    xw_wmma_kernel<<<gGrid, 256, 0, stream>>>(X, W, Y, nNodes, mTiles);

    // 2) out = bias (broadcast)
    const int n = nNodes * OUT_F;
    init_bias_kernel<<<(n + 255) / 256, 256, 0, stream>>>(out, bias, n);

    // 3) Scatter-add all edges: out[row] += val * Y[k][col]
    const long long sThreads = (long long)E * 32;
    dim3 sGrid((unsigned)((sThreads + 255) / 256), NSUP);
    spmm_scatter_kernel<<<sGrid, 256, 0, stream>>>(rows, cols, vals, Y, out,
                                                   E, nNodes);
}